// Summarizer_46462956208271
// MI455X (gfx1250) — compile-verified
//
#include <hip/hip_runtime.h>
#include <math.h>

// Problem constants (match reference):
#define BATCH   128
#define SEQ     512
#define DIM     768
#define MAXC    64

typedef __attribute__((ext_vector_type(2))) float v2f;
typedef __attribute__((ext_vector_type(4))) float v4f;
typedef __attribute__((ext_vector_type(8))) float v8f;

// One block per batch row. 128 threads = 4 wave32's; wave w computes output
// rows m = 16w .. 16w+15 via V_WMMA_F32_16X16X4_F32 (full fp32 precision).
__global__ __launch_bounds__(128) void summarizer_wmma_kernel(
    const float* __restrict__ enc,        // [B, S, D] f32
    const float* __restrict__ W,          // [D] f32 (D x 1)
    const float* __restrict__ bias,       // [1] f32
    const unsigned char* __restrict__ cls,// [B, S] bool (1 byte each)
    float* __restrict__ out)              // [B, MAXC] f32
{
    __shared__ alignas(16) float sW[DIM]; // W staged in LDS (3 KB), 16B aligned
    __shared__ unsigned char  sCls[SEQ];  // flag row
    __shared__ int            sPos[MAXC]; // first 64 true positions
    __shared__ int            sChunkCnt[128];
    __shared__ int            sChunkBase[128];
    __shared__ int            sCount;

    const int b    = blockIdx.x;
    const int tid  = threadIdx.x;
    const int lane = tid & 31;
    const int wave = tid >> 5;

    // ---- Stage W and the cls flag row into LDS ----
    for (int i = tid; i < DIM; i += 128) sW[i] = W[i];
    {
        const unsigned char* row = cls + (size_t)b * SEQ;
        for (int i = tid; i < SEQ; i += 128) sCls[i] = row[i];
    }
    if (tid < MAXC) sPos[tid] = -1;
    __syncthreads();

    // ---- Stable "first 64 true positions": chunk counts + prefix + scatter ----
    const int c0 = tid * 4;                 // each thread owns 4 consecutive positions
    unsigned char f0 = sCls[c0 + 0], f1 = sCls[c0 + 1],
                  f2 = sCls[c0 + 2], f3 = sCls[c0 + 3];
    int localCnt = (f0 ? 1 : 0) + (f1 ? 1 : 0) + (f2 ? 1 : 0) + (f3 ? 1 : 0);
    sChunkCnt[tid] = localCnt;
    __syncthreads();
    if (tid == 0) {
        int acc = 0;
        for (int i = 0; i < 128; ++i) { sChunkBase[i] = acc; acc += sChunkCnt[i]; }
        sCount = acc < MAXC ? acc : MAXC;
    }
    __syncthreads();
    {
        int base = sChunkBase[tid];
        int o = 0;
        if (f0) { if (base + o < MAXC) sPos[base + o] = c0 + 0; ++o; }
        if (f1) { if (base + o < MAXC) sPos[base + o] = c0 + 1; ++o; }
        if (f2) { if (base + o < MAXC) sPos[base + o] = c0 + 2; ++o; }
        if (f3) { if (base + o < MAXC) sPos[base + o] = c0 + 3; ++o; }
    }
    __syncthreads();

    // ---- GEMV tile via f32 WMMA (D = A(16x4) x B(4x16) + C) ----
    // A layout (32-bit, 16x4): lanes 0..15 = M=lane hold {K0,K1};
    //                          lanes 16..31 = M=lane-16 hold {K2,K3}.
    // K-slot assignment is ours as long as A and B agree, so per 8-K step:
    //   WMMA#1 slots = {k,k+1 (lo) | k+4,k+5 (hi)}  -> both halves use .xy
    //   WMMA#2 slots = {k+2,k+3 (lo) | k+6,k+7 (hi)} -> both halves use .zw
    // of one b128 load at (base + k + 4*hiHalf). Every byte loaded exactly once.
    // B column N=0 (lanes 0,16) is all that matters for D column 0; other
    // lanes' B values only pollute D columns 1..15, which we never read.
    const int  mLocal = lane & 15;          // row within this wave's 16-row tile
    const int  m      = wave * 16 + mLocal; // output slot 0..63
    const int  count  = sCount;
    const int  pos    = (m < count) ? sPos[m] : 0;  // fallback row; overridden in epilogue
    const bool hiHalf = (lane >= 16);

    const float* gBase = enc + ((size_t)b * SEQ + (size_t)pos) * DIM + (hiHalf ? 4 : 0);
    const float* wBase = sW + (hiHalf ? 4 : 0);

    v8f acc = {};
    for (int k0 = 0; k0 < DIM; k0 += 64) {
        __builtin_prefetch(gBase + k0 + 64, 0, 3);  // global_prefetch_b8, next chunk (speculative)
        #pragma unroll
        for (int kk = 0; kk < 64; kk += 8) {
            const int k = k0 + kk;
            v4f g = *(const v4f*)(gBase + k);       // global_load_b128
            v4f w = *(const v4f*)(wBase + k);       // ds_load_b128

            v2f a0; a0.x = g.x; a0.y = g.y;
            v2f b0; b0.x = w.x; b0.y = w.y;
            acc = __builtin_amdgcn_wmma_f32_16x16x4_f32(
                false, a0, false, b0, (short)0, acc, false, false);

            v2f a1; a1.x = g.z; a1.y = g.w;
            v2f b1; b1.x = w.z; b1.y = w.w;
            acc = __builtin_amdgcn_wmma_f32_16x16x4_f32(
                false, a1, false, b1, (short)0, acc, false, false);
        }
    }

    // ---- Column N=0 of D lives in lane 0 (M = 0..7) and lane 16 (M = 8..15) ----
    const float bv = bias[0];
    if (mLocal == 0) {                       // lanes 0 and 16
        const int mBase = wave * 16 + (hiHalf ? 8 : 0);
        v4f r0, r1;
        #pragma unroll
        for (int r = 0; r < 4; ++r) {
            // invalid rows -> gathered row is zero in the reference -> sigmoid(bias)
            float x0 = (mBase + r     < count) ? acc[r]     + bv : bv;
            float x1 = (mBase + r + 4 < count) ? acc[r + 4] + bv : bv;
            r0[r] = 1.0f / (1.0f + __expf(-x0));
            r1[r] = 1.0f / (1.0f + __expf(-x1));
        }
        float* o = out + (size_t)b * MAXC + mBase;
        *(v4f*)(o)     = r0;                 // two global_store_b128
        *(v4f*)(o + 4) = r1;
    }
}

extern "C" void kernel_launch(void* const* d_in, const int* in_sizes, int n_in,
                              void* d_out, int out_size, void* d_ws, size_t ws_size,
                              hipStream_t stream) {
    (void)in_sizes; (void)n_in; (void)d_ws; (void)ws_size; (void)out_size;
    const float*         enc  = (const float*)d_in[0];         // [B,S,D]
    const float*         W    = (const float*)d_in[1];         // [D,1]
    const float*         bias = (const float*)d_in[2];         // [1]
    const unsigned char* cls  = (const unsigned char*)d_in[3]; // [B,S] bool
    // d_in[4] = max_num_cls (scalar) — compile-time MAXC used.
    float* out = (float*)d_out;                                // [B,MAXC,1]

    summarizer_wmma_kernel<<<dim3(BATCH), dim3(128), 0, stream>>>(enc, W, bias, cls, out);
}